// Flow_49160195670664
// MI455X (gfx1250) — compile-verified
//
#include <hip/hip_runtime.h>

typedef __attribute__((ext_vector_type(2))) float v2f;
typedef __attribute__((ext_vector_type(8))) float v8f;

#define WMMA_F32(a, b, c) \
  __builtin_amdgcn_wmma_f32_16x16x4_f32(false, (a), false, (b), (short)0, (c), false, false)

static constexpr int Tn = 32768;  // time points
static constexpr int NS = 256;    // state dim
static constexpr int L1 = 32;     // level-1 chunk length
static constexpr int C1 = 1024;   // level-1 chunk count (C1*L1 == Tn)
static constexpr int L2 = 16;     // level-2 chunk length
static constexpr int C2 = 64;     // level-2 chunk count (C2*L2 == C1)
static constexpr int SC_COLS = 16; // chunk-columns per workgroup in scan kernels

// Hermite basis at the 6 Dopri5 stage offsets: H[s][j], j = {h00, h10, h01, h11}
__constant__ float d_H[6][4] = {
  {1.0f, 0.0f, 0.0f, 0.0f},
  {0.896f, 0.128f, 0.104f, -0.032f},
  {0.784f, 0.147f, 0.216f, -0.063f},
  {0.104f, 0.032f, 0.896f, -0.128f},
  {25.0f/729.0f, 8.0f/729.0f, 704.0f/729.0f, -64.0f/729.0f},
  {0.0f, 0.0f, 1.0f, 0.0f},
};

// ---------------------------------------------------------------- elementwise
__global__ void k_scale_dt(float* __restrict__ o, const float* __restrict__ i,
                           int n, const float* __restrict__ t) {
  int idx = blockIdx.x * 256 + threadIdx.x;
  if (idx < n) { float dt = t[1] - t[0]; o[idx] = dt * i[idx]; }
}
__global__ void k_copy(float* __restrict__ o, const float* __restrict__ i, int n) {
  int idx = blockIdx.x * 256 + threadIdx.x;
  if (idx < n) o[idx] = i[idx];
}
__global__ void k_zero(float* __restrict__ o, int n) {
  int idx = blockIdx.x * 256 + threadIdx.x;
  if (idx < n) o[idx] = 0.0f;
}
__global__ void k_eye(float* __restrict__ o) { // 256x256 identity
  int idx = blockIdx.x * 256 + threadIdx.x;
  if (idx < 65536) o[idx] = ((idx >> 8) == (idx & 255)) ? 1.0f : 0.0f;
}
__global__ void k_axpy(float* __restrict__ o, const float* __restrict__ i,
                       int n, float alpha) {
  int idx = blockIdx.x * 256 + threadIdx.x;
  if (idx < n) o[idx] += alpha * i[idx];
}
__global__ void k_addblock(float* __restrict__ Lm, const float* __restrict__ Bd, int blk) {
  int idx = blockIdx.x * 256 + threadIdx.x;   // 256x32 block add into 256x192
  if (idx < 8192) { int r = idx >> 5, c = idx & 31; Lm[r * 192 + blk * 32 + c] += Bd[idx]; }
}
// Qcat[k,r] (k = 32*j + c) = sum_s H[s][j] * W[r, 32*s + c]
__global__ void k_qcat(float* __restrict__ Q, const float* __restrict__ W) {
  int idx = blockIdx.x * 256 + threadIdx.x;   // 128*256
  if (idx >= 128 * 256) return;
  int k = idx >> 8, r = idx & 255;
  int j = k >> 5, c = k & 31;
  float s = 0.0f;
  for (int st = 0; st < 6; ++st) s += d_H[st][j] * W[r * 192 + 32 * st + c];
  Q[idx] = s;
}
// P[n, 0:128] = [u_n, m_n, u_{n+1}, m_{n+1}]  (row 32767 zero-padded)
__global__ void k_buildP(float* __restrict__ P, const float* __restrict__ u) {
  int idx = blockIdx.x * 256 + threadIdx.x;   // Tn*128
  if (idx >= Tn * 128) return;
  int n = idx >> 7, k = idx & 127;
  float val = 0.0f;
  if (n < Tn - 1) {
    int j = k >> 5, c = k & 31;
    int m = n + ((j >= 2) ? 1 : 0);
    if ((j & 1) == 0) {
      val = u[m * 32 + c];
    } else {                                   // slope m_m
      if (m == 0)            val = u[32 + c] - u[c];
      else if (m == Tn - 1)  val = u[(Tn - 1) * 32 + c] - u[(Tn - 2) * 32 + c];
      else                   val = 0.5f * (u[(m + 1) * 32 + c] - u[(m - 1) * 32 + c]);
    }
  }
  P[idx] = val;
}

// ---------------------------------------------------------------- generic GEMM
// D[M,N] = alpha * A[M,K] @ Bop[K,N] + beta * Cin[M,N]; Bop = transB ? B^T : B.
// One 16x16 tile per wave, V_WMMA_F32_16X16X4_F32, K%4==0, M%16==0, N%16==0.
// transB/beta branches hoisted OUT of hot loops; contiguous pairs loaded as v2f.
__global__ __launch_bounds__(128) void gemm_kernel(
    float* __restrict__ D, int ldd,
    const float* __restrict__ A, int lda,
    const float* __restrict__ B, int ldb,
    const float* __restrict__ Cin, int ldc,
    int Mr, int Nc, int Kd, float alpha, float beta, int transB) {
  const int lane = threadIdx.x & 31, wave = threadIdx.x >> 5;
  const int hl = lane >> 4, lr = lane & 15;
  const int tilesN = Nc >> 4;
  const int tile = blockIdx.x * 4 + wave;
  if (tile >= (Mr >> 4) * tilesN) return;
  const int tm = tile / tilesN, tn = tile % tilesN;
  const float* Ar = A + (size_t)(tm * 16 + lr) * lda + 2 * hl;
  v8f acc = {};
  if (transB) {
    const float* Br = B + (size_t)(tn * 16 + lr) * ldb + 2 * hl;
    #pragma unroll 4
    for (int k = 0; k < Kd; k += 4) {
      v2f a = *(const v2f*)(Ar + k);
      v2f b = *(const v2f*)(Br + k);
      acc = WMMA_F32(a, b, acc);
    }
  } else {
    const float* Bc = B + (size_t)(2 * hl) * ldb + tn * 16 + lr;
    #pragma unroll 4
    for (int k = 0; k < Kd; k += 4) {
      v2f a = *(const v2f*)(Ar + k);
      v2f b;
      b[0] = Bc[(size_t)k * ldb];
      b[1] = Bc[(size_t)k * ldb + ldb];
      acc = WMMA_F32(a, b, acc);
    }
  }
  float* Dp = D + (size_t)(tm * 16 + 8 * hl) * ldd + tn * 16 + lr;
  if (beta != 0.0f) {
    const float* Cp = Cin + (size_t)(tm * 16 + 8 * hl) * ldc + tn * 16 + lr;
    #pragma unroll
    for (int v = 0; v < 8; ++v)
      Dp[(size_t)v * ldd] = alpha * acc[v] + beta * Cp[(size_t)v * ldc];
  } else {
    #pragma unroll
    for (int v = 0; v < 8; ++v)
      Dp[(size_t)v * ldd] = alpha * acc[v];
  }
}

// ---------------------------------------------------------------- batched scan
// Each WG owns SC_COLS chunk-columns; per step: X <- M @ X + g[row]. M (256KB)
// lives in LDS (320KB WGP budget), X double-buffered in LDS. 8 waves x 2 row
// tiles each => full 256x16 update per step via 128 chained f32 WMMAs/wave.
__global__ __launch_bounds__(256) void scan_kernel(
    const float* __restrict__ Mg, const float* __restrict__ G,
    const float* __restrict__ Init,      // per-chunk initial state [cols,256] or null(=0)
    float* __restrict__ OutAll,          // write state before each step, ld=256, or null
    float* __restrict__ OutEnd,          // write state after L steps, [cols,256], or null
    int L) {
  extern __shared__ float smem[];
  float* Ms = smem;                       // 256*256
  float* Xbuf0 = smem + 65536;            // 256*SC_COLS
  float* Xbuf1 = Xbuf0 + 256 * SC_COLS;
  const int tid = threadIdx.x;
  const int lane = tid & 31, wave = tid >> 5;
  const int hl = lane >> 4, lr = lane & 15;
  const int cbase = blockIdx.x * SC_COLS;
  const int tr0 = wave * 2;               // this wave's two row tiles

  { // cooperative load of M into LDS
    const float4* src = (const float4*)Mg;
    float4* dst = (float4*)Ms;
    for (int k = tid; k < 65536 / 4; k += 256) dst[k] = src[k];
  }
  for (int e = 0; e < SC_COLS; ++e)       // thread tid is state row r
    Xbuf0[tid * SC_COLS + e] = Init ? Init[(size_t)(cbase + e) * 256 + tid] : 0.0f;
  __syncthreads();

  // this lane's G stream: column (cbase+lr), rows i = 0..L-1
  const float* gbase = G + ((size_t)(cbase + lr) * L) * 256 + tr0 * 16 + 8 * hl;

  float* Xc = Xbuf0;
  float* Xn = Xbuf1;
  for (int i = 0; i < L; ++i) {
    if (OutAll) {
      for (int e = 0; e < SC_COLS; ++e) {
        size_t m = (size_t)(cbase + e) * L + i;
        OutAll[m * 256 + tid] = Xc[tid * SC_COLS + e];
      }
    }
    // accumulators seeded with g for this lane's column
    const float* gp = gbase + (size_t)i * 256;
    if (i + 1 < L) __builtin_prefetch(gp + 256, 0, 1);   // next step's g row
    v8f acc0, acc1;
    {
      float4 ga = *(const float4*)(gp);
      float4 gb = *(const float4*)(gp + 4);
      float4 gc = *(const float4*)(gp + 16);
      float4 gd = *(const float4*)(gp + 20);
      acc0[0]=ga.x; acc0[1]=ga.y; acc0[2]=ga.z; acc0[3]=ga.w;
      acc0[4]=gb.x; acc0[5]=gb.y; acc0[6]=gb.z; acc0[7]=gb.w;
      acc1[0]=gc.x; acc1[1]=gc.y; acc1[2]=gc.z; acc1[3]=gc.w;
      acc1[4]=gd.x; acc1[5]=gd.y; acc1[6]=gd.z; acc1[7]=gd.w;
    }
    const float* Mr0 = Ms + (size_t)(tr0 * 16 + lr) * 256 + 2 * hl;
    const float* Mr1 = Mr0 + 16 * 256;
    const float* Xk = Xc + 2 * hl * SC_COLS + lr;
    #pragma unroll 8
    for (int kc = 0; kc < 64; ++kc) {
      v2f a0 = *(const v2f*)(Mr0 + kc * 4);
      v2f a1 = *(const v2f*)(Mr1 + kc * 4);
      v2f b;
      b[0] = Xk[kc * 4 * SC_COLS];
      b[1] = Xk[kc * 4 * SC_COLS + SC_COLS];
      acc0 = WMMA_F32(a0, b, acc0);
      acc1 = WMMA_F32(a1, b, acc1);
    }
    #pragma unroll
    for (int v = 0; v < 8; ++v) {
      Xn[(tr0 * 16 + 8 * hl + v) * SC_COLS + lr] = acc0[v];
      Xn[((tr0 + 1) * 16 + 8 * hl + v) * SC_COLS + lr] = acc1[v];
    }
    __syncthreads();
    float* tmp = Xc; Xc = Xn; Xn = tmp;
  }
  if (OutEnd) {
    for (int e = 0; e < SC_COLS; ++e)
      OutEnd[(size_t)(cbase + e) * 256 + tid] = Xc[tid * SC_COLS + e];
  }
}

// ---------------------------------------------------------------- serial carry
// t_0 = x0; t_{j+1} = M512 t_j + Z[j]; store all t_j (level-2 chunk inits).
__global__ __launch_bounds__(256) void carry_kernel(
    const float* __restrict__ Mg, const float* __restrict__ Z,
    const float* __restrict__ x0, float* __restrict__ Tcar, int n) {
  __shared__ float tv[256];
  int r = threadIdx.x;
  tv[r] = x0[r];
  Tcar[r] = tv[r];
  __syncthreads();
  const float* Mr = Mg + (size_t)r * 256;
  for (int j = 0; j + 1 < n; ++j) {
    float acc = Z[(size_t)j * 256 + r];
    #pragma unroll 8
    for (int k = 0; k < 256; ++k) acc += Mr[k] * tv[k];
    __syncthreads();
    tv[r] = acc;
    Tcar[(size_t)(j + 1) * 256 + r] = acc;
    __syncthreads();
  }
}

// ---------------------------------------------------------------- host driver
static inline void launch_gemm(hipStream_t s, float* D, int ldd,
                               const float* A, int lda, const float* B, int ldb,
                               const float* Cin, int ldc,
                               int M, int N, int K, float alpha, float beta, int transB) {
  int tiles = (M / 16) * (N / 16);
  gemm_kernel<<<(tiles + 3) / 4, 128, 0, s>>>(D, ldd, A, lda, B, ldb, Cin, ldc,
                                              M, N, K, alpha, beta, transB);
}

extern "C" void kernel_launch(void* const* d_in, const int* in_sizes, int n_in,
                              void* d_out, int out_size, void* d_ws, size_t ws_size,
                              hipStream_t stream) {
  (void)in_sizes; (void)n_in; (void)out_size; (void)ws_size;
  const float* t  = (const float*)d_in[0];
  const float* u  = (const float*)d_in[1];
  const float* x0 = (const float*)d_in[2];
  const float* Am = (const float*)d_in[3];
  const float* Bm = (const float*)d_in[4];
  const float* Cm = (const float*)d_in[5];
  const float* Dm = (const float*)d_in[6];
  float* xout = (float*)d_out;                  // [32768,256]
  float* yout = xout + (size_t)Tn * NS;         // [32768,32]

  float* w = (float*)d_ws;
  float* Ad = w;  w += 65536;                   // dt*A
  float* Bd = w;  w += 8192;                    // dt*B
  float* Kt[6]; for (int i = 0; i < 6; i++) { Kt[i] = w; w += 65536; }
  float* Lt[6]; for (int i = 0; i < 6; i++) { Lt[i] = w; w += 49152; }
  float* TMP  = w; w += 65536;
  float* TMPL = w; w += 49152;
  float* Mm   = w; w += 65536;                  // step matrix
  float* Wm   = w; w += 49152;                  // stage->input weights [256,192]
  float* Qc   = w; w += 32768;                  // Qcat [128,256]
  float* ML   = w; w += 65536;                  // M^32
  float* M512 = w; w += 65536;                  // M^512
  float* PT1  = w; w += 65536;
  float* PT2  = w; w += 65536;
  float* Pm   = w; w += (size_t)Tn * 128;       // Hermite data matrix
  float* G    = w; w += (size_t)Tn * 256;       // per-step drive g_n
  float* Zend = w; w += (size_t)C1 * 256;
  float* Zend2= w; w += (size_t)C2 * 256;
  float* Tcar = w; w += (size_t)C2 * 256;
  float* Scar = w; w += (size_t)C1 * 256;

  auto gdim = [](int n) { return dim3((unsigned)((n + 255) / 256)); };

  // ---- build dt-scaled system + stage matrices (kappa_i = Ktil_i x + Ltil_i u)
  k_scale_dt<<<gdim(65536), 256, 0, stream>>>(Ad, Am, 65536, t);
  k_scale_dt<<<gdim(8192),  256, 0, stream>>>(Bd, Bm, 8192, t);
  k_copy<<<gdim(65536), 256, 0, stream>>>(Kt[0], Ad, 65536);
  k_zero<<<gdim(49152), 256, 0, stream>>>(Lt[0], 49152);
  k_addblock<<<gdim(8192), 256, 0, stream>>>(Lt[0], Bd, 0);

  double aa[7][7] = {};
  aa[2][1] = 0.2;
  aa[3][1] = 3.0/40.0;      aa[3][2] = 9.0/40.0;
  aa[4][1] = 44.0/45.0;     aa[4][2] = -56.0/15.0;     aa[4][3] = 32.0/9.0;
  aa[5][1] = 19372.0/6561.0; aa[5][2] = -25360.0/2187.0;
  aa[5][3] = 64448.0/6561.0; aa[5][4] = -212.0/729.0;
  aa[6][1] = 9017.0/3168.0; aa[6][2] = -355.0/33.0; aa[6][3] = 46732.0/5247.0;
  aa[6][4] = 49.0/176.0;    aa[6][5] = -5103.0/18656.0;
  for (int i = 2; i <= 6; ++i) {
    k_eye<<<gdim(65536), 256, 0, stream>>>(TMP);
    k_zero<<<gdim(49152), 256, 0, stream>>>(TMPL, 49152);
    for (int j = 1; j < i; ++j) {
      float c = (float)aa[i][j];
      k_axpy<<<gdim(65536), 256, 0, stream>>>(TMP,  Kt[j-1], 65536, c);
      k_axpy<<<gdim(49152), 256, 0, stream>>>(TMPL, Lt[j-1], 49152, c);
    }
    launch_gemm(stream, Kt[i-1], 256, Ad, 256, TMP,  256, nullptr, 0, 256, 256, 256, 1.f, 0.f, 0);
    launch_gemm(stream, Lt[i-1], 192, Ad, 256, TMPL, 192, nullptr, 0, 256, 192, 256, 1.f, 0.f, 0);
    k_addblock<<<gdim(8192), 256, 0, stream>>>(Lt[i-1], Bd, i - 1);
  }
  const double bb[7] = {0.0, 35.0/384.0, 0.0, 500.0/1113.0, 125.0/192.0,
                        -2187.0/6784.0, 11.0/84.0};
  k_eye<<<gdim(65536), 256, 0, stream>>>(Mm);
  k_zero<<<gdim(49152), 256, 0, stream>>>(Wm, 49152);
  for (int i = 1; i <= 6; ++i) {
    float c = (float)bb[i];
    if (c != 0.0f) {
      k_axpy<<<gdim(65536), 256, 0, stream>>>(Mm, Kt[i-1], 65536, c);
      k_axpy<<<gdim(49152), 256, 0, stream>>>(Wm, Lt[i-1], 49152, c);
    }
  }
  k_qcat<<<gdim(32768), 256, 0, stream>>>(Qc, Wm);

  // ---- matrix powers: M^32 and M^512 (repeated squaring)
  launch_gemm(stream, PT1, 256, Mm, 256, Mm, 256, nullptr, 0, 256, 256, 256, 1.f, 0.f, 0);   // M^2
  launch_gemm(stream, PT2, 256, PT1, 256, PT1, 256, nullptr, 0, 256, 256, 256, 1.f, 0.f, 0); // M^4
  launch_gemm(stream, PT1, 256, PT2, 256, PT2, 256, nullptr, 0, 256, 256, 256, 1.f, 0.f, 0); // M^8
  launch_gemm(stream, PT2, 256, PT1, 256, PT1, 256, nullptr, 0, 256, 256, 256, 1.f, 0.f, 0); // M^16
  launch_gemm(stream, ML,  256, PT2, 256, PT2, 256, nullptr, 0, 256, 256, 256, 1.f, 0.f, 0); // M^32
  launch_gemm(stream, PT1, 256, ML,  256, ML,  256, nullptr, 0, 256, 256, 256, 1.f, 0.f, 0); // M^64
  launch_gemm(stream, PT2, 256, PT1, 256, PT1, 256, nullptr, 0, 256, 256, 256, 1.f, 0.f, 0); // M^128
  launch_gemm(stream, PT1, 256, PT2, 256, PT2, 256, nullptr, 0, 256, 256, 256, 1.f, 0.f, 0); // M^256
  launch_gemm(stream, M512,256, PT1, 256, PT1, 256, nullptr, 0, 256, 256, 256, 1.f, 0.f, 0); // M^512

  // ---- drive terms: G[T,256] = P[T,128] @ Qcat[128,256]
  k_buildP<<<gdim(Tn * 128), 256, 0, stream>>>(Pm, u);
  launch_gemm(stream, G, 256, Pm, 128, Qc, 256, nullptr, 0, Tn, 256, 128, 1.f, 0.f, 0);

  // ---- hierarchical parallel-in-time scan
  const size_t SCAN_LDS = (size_t)(65536 + 2 * 256 * SC_COLS) * sizeof(float); // 294912 B
  // level-1 phase A: zero-init local scans, 1024 chunks of 32
  scan_kernel<<<C1 / SC_COLS, 256, SCAN_LDS, stream>>>(Mm, G, nullptr, nullptr, Zend, L1);
  // level-2 phase A over the 1024 carries (matrix M^32), 64 chunks of 16
  scan_kernel<<<C2 / SC_COLS, 256, SCAN_LDS, stream>>>(ML, Zend, nullptr, nullptr, Zend2, L2);
  // level-3 serial chain of 64 with M^512, seeded with x0
  carry_kernel<<<1, 256, 0, stream>>>(M512, Zend2, x0, Tcar, C2);
  // level-2 fix-up: emit all 1024 level-1 chunk initial states
  scan_kernel<<<C2 / SC_COLS, 256, SCAN_LDS, stream>>>(ML, Zend, Tcar, Scar, nullptr, L2);
  // level-1 fix-up: emit all x_n rows directly into d_out
  scan_kernel<<<C1 / SC_COLS, 256, SCAN_LDS, stream>>>(Mm, G, Scar, xout, nullptr, L1);

  // ---- outputs: Y = X @ C^T + U @ D^T
  launch_gemm(stream, yout, 32, xout, 256, Cm, 256, nullptr, 0, Tn, 32, 256, 1.f, 0.f, 1);
  launch_gemm(stream, yout, 32, u,    32,  Dm, 32,  yout,   32, Tn, 32, 32,  1.f, 1.f, 1);
}